// Model_76373108457501
// MI455X (gfx1250) — compile-verified
//
#include <hip/hip_runtime.h>
#include <hip/hip_bf16.h>
#include <math.h>

typedef __attribute__((ext_vector_type(2))) float v2f;
typedef __attribute__((ext_vector_type(8))) float v8f;

#define NS 1024
#define N2 (NS * NS)
#define LST 33  // padded LDS row stride (32+1) to avoid bank conflicts

// ---------------------------------------------------------------------------
// WMMA f32 16x16x4 wrapper (CDNA5 fp32 matrix path)
// ---------------------------------------------------------------------------
__device__ __forceinline__ v8f wmma4(v2f a, v2f b, v8f c) {
  return __builtin_amdgcn_wmma_f32_16x16x4_f32(false, a, false, b, (short)0, c,
                                               false, false);
}

// A-tile (16x4, MxK) from row-major [32][LST] matrix.
// lanes 0-15: M=lane,    K = k0+{0,1}
// lanes16-31: M=lane-16, K = k0+{2,3}
__device__ __forceinline__ v2f ldsA(const float* m, int m0, int k0) {
  int l = threadIdx.x & 31;
  int r = m0 + (l & 15);
  int k = k0 + ((l >> 4) << 1);
  v2f a;
  a.x = m[r * LST + k];
  a.y = m[r * LST + k + 1];
  return a;
}
// B-tile (4x16, KxN) from row-major [32][LST] matrix.
__device__ __forceinline__ v2f ldsB(const float* m, int k0, int n0) {
  int l = threadIdx.x & 31;
  int n = n0 + (l & 15);
  int k = k0 + ((l >> 4) << 1);
  v2f b;
  b.x = m[k * LST + n];
  b.y = m[(k + 1) * LST + n];
  return b;
}
// C/D-tile (16x16) store: VGPR v -> (M = v + (lane>=16 ? 8:0), N = lane&15)
__device__ __forceinline__ void ldsStoreC(float* m, int m0, int n0, v8f c) {
  int l = threadIdx.x & 31;
  int n = n0 + (l & 15);
  int mb = m0 + ((l >> 4) << 3);
#pragma unroll
  for (int v = 0; v < 8; ++v) m[(mb + v) * LST + n] = c[v];
}

// ---------------------------------------------------------------------------
// Batched 1024-point complex FFT, one line per wave.
// Four-step: X(32x32), A = F*X, A *= twiddle, Y = A*F, out[k1+32*k2]=Y[k1][k2]
// Complex matmul = 4 real WMMA chains. Direction is a template param so the
// +/-sin operand tables are pure LDS loads (no VALU between LDS and v_wmma).
// ---------------------------------------------------------------------------
template <int INV>
__global__ void __launch_bounds__(128) fft1024_kernel(
    const float* __restrict__ inR, const float* __restrict__ inI,
    float* __restrict__ outR, float* __restrict__ outI, int estride,
    int lstride, float scale) {
  __shared__ float sFc[32 * LST];   // cos
  __shared__ float sFsp[32 * LST];  // +sin
  __shared__ float sFsn[32 * LST];  // -sin
  __shared__ float sX[4][2][32 * LST];

  for (int t = threadIdx.x; t < 1024; t += 128) {
    int r = t >> 5, c = t & 31;
    float ang = 0.19634954084936207f * (float)((r * c) & 31);  // 2pi/32
    float s, cc;
    __sincosf(ang, &s, &cc);
    sFc[r * LST + c] = cc;
    sFsp[r * LST + c] = s;
    sFsn[r * LST + c] = -s;
  }
  __syncthreads();

  // e^{sgn*i*theta}: imag table and its negation, chosen at compile time
  const float* Fi = INV ? sFsp : sFsn;
  const float* Fni = INV ? sFsn : sFsp;

  const int w = threadIdx.x >> 5;
  const int l = threadIdx.x & 31;
  const int line = blockIdx.x * 4 + w;
  const long base = (long)line * lstride;
  float* Xr = sX[w][0];
  float* Xi = sX[w][1];

  // stage 0: gather line into 32x32 LDS matrix X[n1][n2], n = 32*n1 + n2
#pragma unroll 4
  for (int j = 0; j < 32; ++j) {
    long idx = base + (long)(j * 32 + l) * estride;
    Xr[j * LST + l] = inR[idx];
    Xi[j * LST + l] = inI[idx];
  }
  __syncthreads();

  // stage 1: A = F * X  (complex), then twiddle W^(sgn*i*2pi*k1*n2/1024)
  v8f Ar[2][2], Ai[2][2];
#pragma unroll
  for (int ti = 0; ti < 2; ++ti) {
#pragma unroll
    for (int tj = 0; tj < 2; ++tj) {
      v8f cr = {0.f, 0.f, 0.f, 0.f, 0.f, 0.f, 0.f, 0.f};
      v8f ci = {0.f, 0.f, 0.f, 0.f, 0.f, 0.f, 0.f, 0.f};
      for (int k = 0; k < 32; k += 4) {
        v2f fr = ldsA(sFc, ti * 16, k);
        v2f fi = ldsA(Fi, ti * 16, k);
        v2f fni = ldsA(Fni, ti * 16, k);
        v2f br = ldsB(Xr, k, tj * 16);
        v2f bi = ldsB(Xi, k, tj * 16);
        cr = wmma4(fr, br, cr);
        cr = wmma4(fni, bi, cr);
        ci = wmma4(fr, bi, ci);
        ci = wmma4(fi, br, ci);
      }
      int nn = tj * 16 + (l & 15);
      int mb = ti * 16 + ((l >> 4) << 3);
#pragma unroll
      for (int v = 0; v < 8; ++v) {
        int mm = mb + v;
        // signed angle at compile time: sincos(sgn*theta) folds the sign
        float ang = (INV ? 0.006135923151542565f : -0.006135923151542565f) *
                    (float)((mm * nn) & 1023);  // 2pi/1024
        float si, c2;
        __sincosf(ang, &si, &c2);
        float rr = cr[v] * c2 - ci[v] * si;
        float ii = cr[v] * si + ci[v] * c2;
        cr[v] = rr;
        ci[v] = ii;
      }
      Ar[ti][tj] = cr;
      Ai[ti][tj] = ci;
    }
  }
  __syncthreads();
#pragma unroll
  for (int ti = 0; ti < 2; ++ti)
#pragma unroll
    for (int tj = 0; tj < 2; ++tj) {
      ldsStoreC(Xr, ti * 16, tj * 16, Ar[ti][tj]);
      ldsStoreC(Xi, ti * 16, tj * 16, Ai[ti][tj]);
    }
  __syncthreads();

  // stage 3: Y = A * F  (complex)
  v8f Yr[2][2], Yi[2][2];
#pragma unroll
  for (int ti = 0; ti < 2; ++ti) {
#pragma unroll
    for (int tj = 0; tj < 2; ++tj) {
      v8f cr = {0.f, 0.f, 0.f, 0.f, 0.f, 0.f, 0.f, 0.f};
      v8f ci = {0.f, 0.f, 0.f, 0.f, 0.f, 0.f, 0.f, 0.f};
      for (int k = 0; k < 32; k += 4) {
        v2f ar = ldsA(Xr, ti * 16, k);
        v2f ai = ldsA(Xi, ti * 16, k);
        v2f br = ldsB(sFc, k, tj * 16);
        v2f bi = ldsB(Fi, k, tj * 16);
        v2f bni = ldsB(Fni, k, tj * 16);
        cr = wmma4(ar, br, cr);
        cr = wmma4(ai, bni, cr);
        ci = wmma4(ar, bi, ci);
        ci = wmma4(ai, br, ci);
      }
      Yr[ti][tj] = cr;
      Yi[ti][tj] = ci;
    }
  }
  __syncthreads();
#pragma unroll
  for (int ti = 0; ti < 2; ++ti)
#pragma unroll
    for (int tj = 0; tj < 2; ++tj) {
      ldsStoreC(Xr, ti * 16, tj * 16, Yr[ti][tj]);
      ldsStoreC(Xi, ti * 16, tj * 16, Yi[ti][tj]);
    }
  __syncthreads();

  // scatter: out[k1 + 32*k2] = Y[k1][k2]; lane l covers k1=l, k2=j
#pragma unroll 4
  for (int j = 0; j < 32; ++j) {
    long idx = base + (long)(j * 32 + l) * estride;
    outR[idx] = scale * Xr[l * LST + j];
    outI[idx] = scale * Xi[l * LST + j];
  }
}

// ---------------------------------------------------------------------------
// Pointwise / reduction kernels
// ---------------------------------------------------------------------------
#define KWAVE 12319.971190548208f  // 2*pi / 0.00051

__global__ void init_kernel(float* out, float* slots) {
  int t = threadIdx.x;
  if (t < 55) out[t] = 0.f;
  if (t < 100) slots[t] = 0.f;
}

__global__ void lens_kernel(const float* __restrict__ rlist,
                            const float* __restrict__ xpos,
                            const float* __restrict__ ypos,
                            float* __restrict__ phi, float* __restrict__ aper) {
  int idx = blockIdx.x * 256 + threadIdx.x;
  if (idx >= N2) return;
  int i = idx >> 10, j = idx & 1023;
  const float step = 3.6f / 1023.f;
  float x = -1.8f + j * step;
  float y = -1.8f + i * step;
  float T = 0.f;
  int cnt = 0;
  for (int t = 0; t < 29; ++t) {
    float dx = x - xpos[t], dy = y - ypos[t];
    float d2 = dx * dx + dy * dy;
    bool in = d2 <= 0.165f * 0.165f;
    float r = rlist[t];
    float r2s = in ? d2 : 0.f;
    float sag = r - sqrtf(fmaxf(r * r - r2s, 1e-12f));
    if (in) {
      T += sag;
      cnt++;
    }
  }
  phi[idx] = KWAVE * (1.515f - 1.f) * T;
  aper[idx] = (cnt > 0 && (x * x + y * y) <= 0.81f) ? 1.f : 0.f;
}

__global__ void u1_kernel(const float* __restrict__ aper,
                          const float* __restrict__ phi,
                          const float* __restrict__ zlist, int zi,
                          float* __restrict__ Ur, float* __restrict__ Ui) {
  int idx = blockIdx.x * 256 + threadIdx.x;
  if (idx >= N2) return;
  int i = idx >> 10, j = idx & 1023;
  const float step = 3.6f / 1023.f;
  float x = -1.8f + j * step;
  float y = -1.8f + i * step;
  float z = zlist[zi];
  float pin = KWAVE * (x * x + y * y) / (2.f * z);
  float a = aper[idx];
  float s, c;
  __sincosf(pin + phi[idx], &s, &c);
  Ur[idx] = a * c;
  Ui[idx] = a * s;
}

__global__ void mulH_kernel(float* __restrict__ Ur, float* __restrict__ Ui) {
  int idx = blockIdx.x * 256 + threadIdx.x;
  if (idx >= N2) return;
  int a = idx >> 10, b = idx & 1023;
  int ii = (a + 512) & 1023;  // ifftshift index mapping
  int jj = (b + 512) & 1023;
  const float fmaxv = 1024.f / 7.2f;
  const float df = 2.f * fmaxv / 1023.f;
  float fy = -fmaxv + ii * df;
  float fx = -fmaxv + jj * df;
  const float il2 = (1.f / 0.00051f) * (1.f / 0.00051f);
  float kz = sqrtf(fmaxf(il2 - fx * fx - fy * fy, 0.f));
  float ang = 6.2831853071795862f * 8.74f * kz;
  float s, c;
  __sincosf(ang, &s, &c);
  float r = Ur[idx], im = Ui[idx];
  Ur[idx] = r * c - im * s;
  Ui[idx] = r * s + im * c;
}

__global__ void psf_kernel(const float* __restrict__ Ur,
                           const float* __restrict__ Ui, float* __restrict__ p,
                           float* sumslot) {
  __shared__ float sm[256];
  int idx = blockIdx.x * 256 + threadIdx.x;
  float v = 0.f;
  if (idx < N2) {
    float r = Ur[idx], im = Ui[idx];
    v = r * r + im * im;
    p[idx] = v;
  }
  sm[threadIdx.x] = v;
  __syncthreads();
  for (int s = 128; s > 0; s >>= 1) {
    if (threadIdx.x < s) sm[threadIdx.x] += sm[threadIdx.x + s];
    __syncthreads();
  }
  if (threadIdx.x == 0) atomicAdd(sumslot, sm[0]);
}

__global__ void norm_kernel(const float* __restrict__ p, const float* sumslot,
                            float* __restrict__ Ur, float* __restrict__ Ui) {
  int idx = blockIdx.x * 256 + threadIdx.x;
  if (idx >= N2) return;
  float inv = 1.f / *sumslot;
  Ur[idx] = p[idx] * inv;
  Ui[idx] = 0.f;
}

__global__ void diag_kernel(const float* __restrict__ Sr,
                            const float* __restrict__ Si,
                            const float* __restrict__ tF,
                            const float* __restrict__ mask, float* out) {
  __shared__ float sm[256];
  int z = blockIdx.y;
  int idx = blockIdx.x * 256 + threadIdx.x;
  float t = 0.f;
  if (idx < N2) {
    float r = Sr[(size_t)z * N2 + idx], im = Si[(size_t)z * N2 + idx];
    float d = (r * r + im * im - tF[idx]) * (1.f / 1048576.f);
    t = mask[idx] * d * d;
  }
  sm[threadIdx.x] = t;
  __syncthreads();
  for (int s = 128; s > 0; s >>= 1) {
    if (threadIdx.x < s) sm[threadIdx.x] += sm[threadIdx.x + s];
    __syncthreads();
  }
  if (threadIdx.x == 0) atomicAdd(&out[z], sm[0]);
}

__global__ void cross_kernel(const float* __restrict__ S1r,
                             const float* __restrict__ S1i,
                             const float* __restrict__ S2r,
                             const float* __restrict__ S2i,
                             float* __restrict__ Cr, float* __restrict__ Ci) {
  int idx = blockIdx.x * 256 + threadIdx.x;
  if (idx >= N2) return;
  float a = S1r[idx], b = S1i[idx], c = S2r[idx], d = S2i[idx];
  Cr[idx] = a * c + b * d;  // conj(S1)*S2
  Ci[idx] = a * d - b * c;
}

__global__ void absmax_kernel(const float* __restrict__ Cr,
                              const float* __restrict__ Ci,
                              unsigned* maxslot) {
  __shared__ float sm[256];
  int idx = blockIdx.x * 256 + threadIdx.x;
  float v = 0.f;
  if (idx < N2) {
    float r = Cr[idx], im = Ci[idx];
    v = sqrtf(r * r + im * im);
  }
  sm[threadIdx.x] = v;
  __syncthreads();
  for (int s = 128; s > 0; s >>= 1) {
    if (threadIdx.x < s)
      sm[threadIdx.x] = fmaxf(sm[threadIdx.x], sm[threadIdx.x + s]);
    __syncthreads();
  }
  if (threadIdx.x == 0) atomicMax(maxslot, __float_as_uint(sm[0]));
}

__global__ void sumexp_kernel(const float* __restrict__ Cr,
                              const float* __restrict__ Ci,
                              const unsigned* maxslot, float* sumslot) {
  __shared__ float sm[256];
  float vmax = __uint_as_float(*maxslot);
  int idx = blockIdx.x * 256 + threadIdx.x;
  float e = 0.f;
  if (idx < N2) {
    float r = Cr[idx], im = Ci[idx];
    float v = sqrtf(r * r + im * im);
    e = __expf((v - vmax) * 100.f);  // 1/LSE_P
  }
  sm[threadIdx.x] = e;
  __syncthreads();
  for (int s = 128; s > 0; s >>= 1) {
    if (threadIdx.x < s) sm[threadIdx.x] += sm[threadIdx.x + s];
    __syncthreads();
  }
  if (threadIdx.x == 0) atomicAdd(sumslot, sm[0]);
}

__global__ void finalize_kernel(const unsigned* maxslot, const float* sumslot,
                                float* out) {
  int p = threadIdx.x;
  if (p < 45)
    out[10 + p] =
        30.f * 0.01f * (logf(sumslot[p]) + __uint_as_float(maxslot[p]) * 100.f);
}

// ---------------------------------------------------------------------------
// Host orchestration
// ---------------------------------------------------------------------------
extern "C" void kernel_launch(void* const* d_in, const int* in_sizes, int n_in,
                              void* d_out, int out_size, void* d_ws,
                              size_t ws_size, hipStream_t stream) {
  const float* rlist = (const float*)d_in[0];
  const float* xpos = (const float*)d_in[1];
  const float* ypos = (const float*)d_in[2];
  const float* zlist = (const float*)d_in[3];
  const float* tF = (const float*)d_in[4];
  const float* mask = (const float*)d_in[5];
  float* out = (float*)d_out;

  float* ws = (float*)d_ws;
  size_t o = 0;
  float* aper = ws + o; o += N2;
  float* phi  = ws + o; o += N2;
  float* T1r  = ws + o; o += N2;
  float* T1i  = ws + o; o += N2;
  float* T2r  = ws + o; o += N2;
  float* T2i  = ws + o; o += N2;
  float* Sr   = ws + o; o += (size_t)10 * N2;
  float* Si   = ws + o; o += (size_t)10 * N2;
  float* slots = ws + o;  // [0,10): z-sums, [10,55): pair max, [55,100): pair sumexp
  float* zsum = slots;
  unsigned* pmax = (unsigned*)(slots + 10);
  float* psum = slots + 55;

  dim3 pwG(N2 / 256), pwB(256);

  init_kernel<<<1, 128, 0, stream>>>(out, slots);
  lens_kernel<<<pwG, pwB, 0, stream>>>(rlist, xpos, ypos, phi, aper);

  auto fftF = [&](const float* ir, const float* ii, float* orr, float* oi,
                  int estride, int lstride, float scale) {
    fft1024_kernel<0><<<256, 128, 0, stream>>>(ir, ii, orr, oi, estride,
                                               lstride, scale);
  };
  auto fftI = [&](const float* ir, const float* ii, float* orr, float* oi,
                  int estride, int lstride, float scale) {
    fft1024_kernel<1><<<256, 128, 0, stream>>>(ir, ii, orr, oi, estride,
                                               lstride, scale);
  };

  const float invN = 1.f / 1024.f;
  for (int z = 0; z < 10; ++z) {
    u1_kernel<<<pwG, pwB, 0, stream>>>(aper, phi, zlist, z, T1r, T1i);
    fftF(T1r, T1i, T2r, T2i, 1, NS, 1.f);    // fwd rows
    fftF(T2r, T2i, T1r, T1i, NS, 1, 1.f);    // fwd cols
    mulH_kernel<<<pwG, pwB, 0, stream>>>(T1r, T1i);
    fftI(T1r, T1i, T2r, T2i, 1, NS, invN);   // inv rows
    fftI(T2r, T2i, T1r, T1i, NS, 1, invN);   // inv cols
    psf_kernel<<<pwG, pwB, 0, stream>>>(T1r, T1i, T2r, &zsum[z]);
    norm_kernel<<<pwG, pwB, 0, stream>>>(T2r, &zsum[z], T1r, T1i);
    // fftshift dropped: its (-1)^(u+v) sign cancels in |S|^2 and conj(Si)Sj
    fftF(T1r, T1i, T2r, T2i, 1, NS, 1.f);
    fftF(T2r, T2i, Sr + (size_t)z * N2, Si + (size_t)z * N2, NS, 1, 1.f);
  }

  diag_kernel<<<dim3(N2 / 256, 10), pwB, 0, stream>>>(Sr, Si, tF, mask, out);

  int p = 0;
  for (int i1 = 0; i1 < 10; ++i1) {
    for (int i2 = i1 + 1; i2 < 10; ++i2, ++p) {
      cross_kernel<<<pwG, pwB, 0, stream>>>(
          Sr + (size_t)i1 * N2, Si + (size_t)i1 * N2, Sr + (size_t)i2 * N2,
          Si + (size_t)i2 * N2, T1r, T1i);
      fftI(T1r, T1i, T2r, T2i, 1, NS, invN);
      fftI(T2r, T2i, T1r, T1i, NS, 1, invN);
      absmax_kernel<<<pwG, pwB, 0, stream>>>(T1r, T1i, &pmax[p]);
      sumexp_kernel<<<pwG, pwB, 0, stream>>>(T1r, T1i, &pmax[p], &psum[p]);
    }
  }
  finalize_kernel<<<1, 64, 0, stream>>>(pmax, psum, out);
}